// VirtualGNN_80169859547688
// MI455X (gfx1250) — compile-verified
//
#include <hip/hip_runtime.h>

#define NR 50000
#define NV 2000
#define DD 128
#define ERR_N 800000
#define LN 3
#define GN 32

typedef __attribute__((ext_vector_type(2))) float v2f;
typedef __attribute__((ext_vector_type(8))) float v8f;

// ---------------------------------------------------------------------------
// WMMA f32 GEMM:  Out[M x N] = A[M x 128] * W[128 x N] + bias[N]
// One wave computes a 16x64 tile: four 16x16 WMMA accumulator chains sharing
// the A fragment (4x A reuse, 4 independent WMMA dependency chains).
// N is a compile-time constant -> all loop loads/stores use immediate offsets.
// Requires: M % 16 == 0, N % 64 == 0. K fixed at 128 (lda=128, ldw=ldo=N).
// ---------------------------------------------------------------------------
template <int N>
__global__ __launch_bounds__(256) void gemm_wmma_f32(
    const float* __restrict__ A, const float* __restrict__ W,
    const float* __restrict__ bias, float* __restrict__ Out, int M)
{
    constexpr int TN = N / 64;
    int wave = blockIdx.x * (blockDim.x >> 5) + (threadIdx.x >> 5);
    int tilesM = M >> 4;
    if (wave >= tilesM * TN) return;     // uniform per wave -> EXEC stays full
    int tm = wave / TN;
    int tn = wave - tm * TN;
    int m0 = tm << 4, n0 = tn << 6;

    int lane = threadIdx.x & 31;
    int h = lane >> 4;        // half-wave select: K sub-pair
    int q = lane & 15;        // A row / B,D column within tile

    // A 16x4 fragment: lane(h,q) holds A[m0+q][k+2h], A[m0+q][k+2h+1]
    const float* aRow = A + (size_t)(m0 + q) * DD + 2 * h;
    // B 4x16 fragments: lane(h,q) holds W[k+2h][n], W[k+2h+1][n]
    const float* bCol = W + (size_t)(2 * h) * N + n0 + q;

    v8f acc0 = {0.f, 0.f, 0.f, 0.f, 0.f, 0.f, 0.f, 0.f};
    v8f acc1 = acc0, acc2 = acc0, acc3 = acc0;

#pragma unroll
    for (int k = 0; k < DD; k += 4) {
        float2 a2 = *(const float2*)(aRow + k);
        v2f a; a.x = a2.x; a.y = a2.y;
        const float* bp = bCol + k * N;          // k*N is a compile-time constant
        v2f b0; b0.x = bp[0];  b0.y = bp[N];
        v2f b1; b1.x = bp[16]; b1.y = bp[N + 16];
        v2f b2; b2.x = bp[32]; b2.y = bp[N + 32];
        v2f b3; b3.x = bp[48]; b3.y = bp[N + 48];
        acc0 = __builtin_amdgcn_wmma_f32_16x16x4_f32(
            false, a, false, b0, (short)0, acc0, false, false);
        acc1 = __builtin_amdgcn_wmma_f32_16x16x4_f32(
            false, a, false, b1, (short)0, acc1, false, false);
        acc2 = __builtin_amdgcn_wmma_f32_16x16x4_f32(
            false, a, false, b2, (short)0, acc2, false, false);
        acc3 = __builtin_amdgcn_wmma_f32_16x16x4_f32(
            false, a, false, b3, (short)0, acc3, false, false);
    }

    int col0 = n0 + q;
    float bb0 = bias[col0];
    float bb1 = bias[col0 + 16];
    float bb2 = bias[col0 + 32];
    float bb3 = bias[col0 + 48];
    // D layout: lane<16 -> M = r, lane>=16 -> M = r+8
    float* orow = Out + (size_t)(m0 + 8 * h) * N + col0;
#pragma unroll
    for (int r = 0; r < 8; ++r) {
        orow[r * N + 0]  = acc0[r] + bb0;
        orow[r * N + 16] = acc1[r] + bb1;
        orow[r * N + 32] = acc2[r] + bb2;
        orow[r * N + 48] = acc3[r] + bb3;
    }
}

// ---------------------------------------------------------------------------
// Pack layer weights into contiguous slabs:
//  Wcat1 (128x384) = [Wr0+Wr2 | Wn0 | Wn1], bias1(384) = [b0+b2 | 0 | 0]
//  Wcat2 (128x256) = [Wr1 | Wn2],           bias2(256) = [b1 | 0]
// ---------------------------------------------------------------------------
__global__ __launch_bounds__(256) void pack_weights(
    const float* __restrict__ Wr, const float* __restrict__ Wn,
    const float* __restrict__ bs, int layer,
    float* __restrict__ Wcat1, float* __restrict__ bias1,
    float* __restrict__ Wcat2, float* __restrict__ bias2)
{
    int idx = blockIdx.x * blockDim.x + threadIdx.x;
    const float* wr = Wr + (size_t)layer * 3 * DD * DD;
    const float* wn = Wn + (size_t)layer * 3 * DD * DD;
    const float* bb = bs + (size_t)layer * 3 * DD;
    if (idx < DD * 384) {
        int k = idx / 384, n = idx - k * 384;
        int blk = n >> 7, c = n & 127;
        float v;
        if (blk == 0)      v = wr[0 * DD * DD + k * DD + c] + wr[2 * DD * DD + k * DD + c];
        else if (blk == 1) v = wn[0 * DD * DD + k * DD + c];
        else               v = wn[1 * DD * DD + k * DD + c];
        Wcat1[idx] = v;
    }
    if (idx < DD * 256) {
        int k = idx / 256, n = idx - k * 256;
        int blk = n >> 7, c = n & 127;
        Wcat2[idx] = (blk == 0) ? wr[1 * DD * DD + k * DD + c]
                                : wn[2 * DD * DD + k * DD + c];
    }
    if (idx < 384) bias1[idx] = (idx < 128) ? (bb[idx] + bb[2 * DD + idx]) : 0.f;
    if (idx < 256) bias2[idx] = (idx < 128) ? bb[DD + idx] : 0.f;
}

// rr edges: OutR[dst][0:128] += OutR[src][128:256]  (wave per edge, float4/lane)
__global__ __launch_bounds__(256) void scatter_rr(
    float* __restrict__ OutR, const int* __restrict__ src,
    const int* __restrict__ dst, int E)
{
    int w = blockIdx.x * (blockDim.x >> 5) + (threadIdx.x >> 5);
    if (w >= E) return;
    int lane = threadIdx.x & 31;
    int s = src[w], d = dst[w];
    float4 v = *(const float4*)(OutR + (size_t)s * 384 + 128 + lane * 4);
    float* pd = OutR + (size_t)d * 384 + lane * 4;
    atomicAdd(pd + 0, v.x);
    atomicAdd(pd + 1, v.y);
    atomicAdd(pd + 2, v.z);
    atomicAdd(pd + 3, v.w);
}

// r->v: OutV[dst_rv[i]][0:128] += OutR[i][256:384]
__global__ __launch_bounds__(256) void scatter_rv(
    const float* __restrict__ OutR, const int* __restrict__ dstrv,
    float* __restrict__ OutV, int n)
{
    int w = blockIdx.x * (blockDim.x >> 5) + (threadIdx.x >> 5);
    if (w >= n) return;
    int lane = threadIdx.x & 31;
    int d = dstrv[w];
    float4 v = *(const float4*)(OutR + (size_t)w * 384 + 256 + lane * 4);
    float* pd = OutV + (size_t)d * 256 + lane * 4;
    atomicAdd(pd + 0, v.x);
    atomicAdd(pd + 1, v.y);
    atomicAdd(pd + 2, v.z);
    atomicAdd(pd + 3, v.w);
}

// x_r_new = relu(OutR[i][0:128] + OutV[dst_rv[i]][128:256])
__global__ __launch_bounds__(256) void finalize_real(
    const float* __restrict__ OutR, const float* __restrict__ OutV,
    const int* __restrict__ dstrv, float* __restrict__ Xout, int n)
{
    int t = blockIdx.x * blockDim.x + threadIdx.x;
    if (t >= n * 32) return;
    int i = t >> 5, c = (t & 31) << 2;
    int dv = dstrv[i];
    float4 a = *(const float4*)(OutR + (size_t)i * 384 + c);
    float4 b = *(const float4*)(OutV + (size_t)dv * 256 + 128 + c);
    float4 r;
    r.x = fmaxf(a.x + b.x, 0.f);
    r.y = fmaxf(a.y + b.y, 0.f);
    r.z = fmaxf(a.z + b.z, 0.f);
    r.w = fmaxf(a.w + b.w, 0.f);
    *(float4*)(Xout + (size_t)i * DD + c) = r;
}

// x_v_new = relu(OutV[j][0:128])
__global__ __launch_bounds__(256) void finalize_virt(
    const float* __restrict__ OutV, float* __restrict__ Xout, int n)
{
    int t = blockIdx.x * blockDim.x + threadIdx.x;
    if (t >= n * 32) return;
    int j = t >> 5, c = (t & 31) << 2;
    float4 a = *(const float4*)(OutV + (size_t)j * 256 + c);
    float4 r;
    r.x = fmaxf(a.x, 0.f);
    r.y = fmaxf(a.y, 0.f);
    r.z = fmaxf(a.z, 0.f);
    r.w = fmaxf(a.w, 0.f);
    *(float4*)(Xout + (size_t)j * DD + c) = r;
}

__global__ __launch_bounds__(256) void zero_f(float* __restrict__ p, int n)
{
    int t = blockIdx.x * blockDim.x + threadIdx.x;
    if (t < n) p[t] = 0.f;
}

__global__ __launch_bounds__(256) void count_groups(
    const int* __restrict__ batch, float* __restrict__ cnt, int n)
{
    int t = blockIdx.x * blockDim.x + threadIdx.x;
    if (t < n) atomicAdd(cnt + batch[t], 1.0f);
}

__global__ __launch_bounds__(256) void accum_graph(
    const float* __restrict__ X, const int* __restrict__ batch,
    float* __restrict__ acc, int n)
{
    int t = blockIdx.x * blockDim.x + threadIdx.x;
    if (t >= n * 32) return;
    int i = t >> 5, c = (t & 31) << 2;
    int g = batch[i];
    float4 v = *(const float4*)(X + (size_t)i * DD + c);
    float* pa = acc + (size_t)g * DD + c;
    atomicAdd(pa + 0, v.x);
    atomicAdd(pa + 1, v.y);
    atomicAdd(pa + 2, v.z);
    atomicAdd(pa + 3, v.w);
}

__global__ __launch_bounds__(256) void graph_final(
    const float* __restrict__ acc, const float* __restrict__ cnt,
    float* __restrict__ out)
{
    int t = blockIdx.x * blockDim.x + threadIdx.x;
    if (t >= GN * DD) return;
    int g = t >> 7;
    out[t] = acc[t] / fmaxf(cnt[g], 1.0f);
}

extern "C" void kernel_launch(void* const* d_in, const int* in_sizes, int n_in,
                              void* d_out, int out_size, void* d_ws, size_t ws_size,
                              hipStream_t stream)
{
    (void)in_sizes; (void)n_in; (void)out_size; (void)ws_size;
    const float* x_real = (const float*)d_in[0];
    const float* x_virt = (const float*)d_in[1];
    const float* W_root = (const float*)d_in[2];
    const float* W_nbr  = (const float*)d_in[3];
    const float* bias   = (const float*)d_in[4];
    const int* edge_rr  = (const int*)d_in[5];
    const int* edge_rv  = (const int*)d_in[6];
    const int* batch    = (const int*)d_in[8];
    const int* src_rr = edge_rr;
    const int* dst_rr = edge_rr + ERR_N;
    const int* dst_rv = edge_rv + NR;     // edge_rv row1 = virtual destination per real node

    char* wp = (char*)d_ws;
    auto alloc = [&](size_t bytes) -> float* {
        float* r = (float*)wp;
        wp += (bytes + 255) & ~(size_t)255;
        return r;
    };
    float* XrA   = alloc((size_t)NR * DD * 4);
    float* XrB   = alloc((size_t)NR * DD * 4);
    float* XvA   = alloc((size_t)NV * DD * 4);
    float* XvB   = alloc((size_t)NV * DD * 4);
    float* OutR  = alloc((size_t)NR * 384 * 4);
    float* OutV  = alloc((size_t)NV * 256 * 4);
    float* Wcat1 = alloc((size_t)DD * 384 * 4);
    float* bias1 = alloc(384 * 4);
    float* Wcat2 = alloc((size_t)DD * 256 * 4);
    float* bias2 = alloc(256 * 4);
    float* acc   = alloc((size_t)(GN * DD + GN) * 4);   // 32x128 sums + 32 counts
    float* cnt   = acc + GN * DD;

    float* out_node  = (float*)d_out;                    // 50000 x 128
    float* out_graph = (float*)d_out + (size_t)NR * DD;  // 32 x 128

    const float* cur_xr = x_real;
    const float* cur_xv = x_virt;

    for (int l = 0; l < LN; ++l) {
        pack_weights<<<(DD * 384 + 255) / 256, 256, 0, stream>>>(
            W_root, W_nbr, bias, l, Wcat1, bias1, Wcat2, bias2);

        {   // real-node GEMM: root(+fused bias) | rr-messages | rv-messages
            int waves = (NR / 16) * (384 / 64);
            gemm_wmma_f32<384><<<(waves + 7) / 8, 256, 0, stream>>>(
                cur_xr, Wcat1, bias1, OutR, NR);
        }
        {   // virtual-node GEMM: root(+bias) | vr-messages
            int waves = (NV / 16) * (256 / 64);
            gemm_wmma_f32<256><<<(waves + 7) / 8, 256, 0, stream>>>(
                cur_xv, Wcat2, bias2, OutV, NV);
        }

        scatter_rr<<<(ERR_N + 7) / 8, 256, 0, stream>>>(OutR, src_rr, dst_rr, ERR_N);
        if (l < LN - 1)
            scatter_rv<<<(NR + 7) / 8, 256, 0, stream>>>(OutR, dst_rv, OutV, NR);

        float* xr_out = (l == 0) ? XrA : (l == 1) ? XrB : out_node;
        finalize_real<<<(NR * 32 + 255) / 256, 256, 0, stream>>>(
            OutR, OutV, dst_rv, xr_out, NR);
        if (l < LN - 1) {
            float* xv_out = (l == 0) ? XvA : XvB;
            finalize_virt<<<(NV * 32 + 255) / 256, 256, 0, stream>>>(OutV, xv_out, NV);
            cur_xv = xv_out;
        }
        cur_xr = xr_out;
    }

    // mean readout over real nodes per graph
    zero_f<<<(GN * DD + GN + 255) / 256, 256, 0, stream>>>(acc, GN * DD + GN);
    count_groups<<<(NR + 255) / 256, 256, 0, stream>>>(batch, cnt, NR);
    accum_graph<<<(NR * 32 + 255) / 256, 256, 0, stream>>>(cur_xr, batch, acc, NR);
    graph_final<<<(GN * DD + 255) / 256, 256, 0, stream>>>(acc, cnt, out_graph);
}